// OTWordAttention_4655744549008
// MI455X (gfx1250) — compile-verified
//
#include <hip/hip_runtime.h>
#include <hip/hip_bf16.h>
#include <math.h>

// Problem constants (match reference)
#define BATCH 2048
#define NWORD 64
#define DIM   768
#define KP    8
#define EPSI  1e-5f
#define NIT   5
#define XSTR  772   // LDS row stride in floats: 772 % 64 == 4 -> conflict-free strided reads

typedef __attribute__((ext_vector_type(2))) float v2f;
typedef __attribute__((ext_vector_type(8))) float v8f;
typedef __attribute__((ext_vector_type(4))) unsigned int v4u;
typedef __attribute__((ext_vector_type(8))) int v8i;
typedef __attribute__((ext_vector_type(4))) int v4i;

// branch-free tanh: 1 - 2/(e^{2x}+1); saturates correctly via inf/zero limits
__device__ __forceinline__ float fast_tanh(float x) {
    float e = __expf(2.0f * x);
    return 1.0f - __fdividef(2.0f, e + 1.0f);
}

__launch_bounds__(256)
__global__ void ot_word_attn_fused(const float* __restrict__ X,   // [B, N, D]
                                   const float* __restrict__ W,   // [D, D] (row = out feature)
                                   const float* __restrict__ Bv,  // [D]
                                   const float* __restrict__ P,   // [K, D]
                                   float* __restrict__ Out)       // s [B,D] ++ attn [B,N]
{
    __shared__ float xs[NWORD * XSTR];   // x tile, resident for GEMM-A and pooling (193 KB)
    __shared__ float wt[32 * XSTR];      // two W column tiles (32 out-features x 768, 96.5 KB)
    __shared__ float bl[DIM];            // bias
    __shared__ float costs[NWORD * KP];  // cost / log-plan scratch
    __shared__ float colL[KP];
    __shared__ float attnl[NWORD];

    const int b    = blockIdx.x;
    const int tid  = threadIdx.x;
    const int lane = tid & 31;
    const int wv   = tid >> 5;      // wave 0..7
    const int rt   = wv & 3;        // row tile 0..3
    const int co   = wv >> 2;       // column-tile subgroup 0/1
    const int half = lane >> 4;     // 0: lanes 0-15, 1: lanes 16-31
    const int lm   = lane & 15;

    const float* xb = X + (size_t)b * NWORD * DIM;

    // ---- x tile (64x768) -> LDS via Tensor Data Mover, iterate mode ----
    // One row (768 f32) per iteration; LDS stride 772 floats reproduces the
    // padded, bank-conflict-free layout. Issued by wave 0 only (TDM is a
    // per-wave DMA op; EXEC ignored), tracked with TENSORcnt.
    if (wv == 0) {
        const unsigned long long ga = (unsigned long long)(uintptr_t)xb;
        v4u g0;
        g0.x = 1u;                                   // count=1 valid descriptor
        g0.y = (unsigned int)(uintptr_t)xs;          // lds_addr (bytes, LDS aperture low 32b)
        g0.z = (unsigned int)ga;                     // global_addr[31:0]
        g0.w = (unsigned int)((ga >> 32) & 0x01FFFFFFu) | (2u << 30);  // addr[56:32] | type=2

        v8i g1;
        g1[0] = (2 << 16) | (1 << 19);               // data_size=4B, iterate_enable=1
        g1[1] = (int)(768u << 16);                   // tensor_dim0[15:0]=768 (bits 63:48)
        g1[2] = (int)(64u << 16);                    // tensor_dim1[15:0]=64  (bits 95:80)
        g1[3] = (int)(768u << 16);                   // tile_dim0=768         (bits 127:112)
        g1[4] = 1;                                   // tile_dim1=1, tile_dim2=0
        g1[5] = 768;                                 // tensor_dim0_stride[31:0]
        g1[6] = 0;                                   // stride hi / dim1_stride lo
        g1[7] = 0;

        v4i g2;
        g2[0] = 1;                                   // tensor_dim2
        g2[1] = XSTR;                                // lds_addr_increment (data_size units)
        g2[2] = DIM;                                 // global_addr_increment[31:0]
        g2[3] = (63 << 16);                          // iterate_count=63 (64 rows)

        v4i g3 = {0, 0, 0, 0};
        v8i g4 = {0, 0, 0, 0, 0, 0, 0, 0};           // trailing group (clang-23 6-arg form)
        __builtin_amdgcn_tensor_load_to_lds(g0, g1, g2, g3, g4, 0);
    }

    // bias -> LDS (overlaps with the TDM transfer)
    for (int i = tid; i < DIM / 4; i += 256) {
        *(float4*)(&bl[i * 4]) = *(const float4*)(Bv + (size_t)i * 4);
    }

    if (wv == 0) {
        __builtin_amdgcn_s_wait_tensorcnt(0);        // x tile landed in LDS
    }

    // per-lane squared-distance accumulators: rows j (within half), prototypes k
    float ca[8][KP];
#pragma unroll
    for (int j = 0; j < 8; ++j)
#pragma unroll
        for (int k = 0; k < KP; ++k) ca[j][k] = 0.0f;

    const int arow = rt * 16 + lm;  // A-operand row for this lane

    // ---- fused GEMM(tanh) + cost accumulation: 48 col tiles, 2 per iteration ----
    for (int c = 0; c < DIM / 32; ++c) {
        const int ecbase = c * 32;          // W rows [ecbase, ecbase+32) staged
        const int ec     = ecbase + co * 16; // this wave-group's output columns
        __syncthreads();  // previous wt consumers done; (c==0) publishes TDM x tile
        for (int i = tid; i < 32 * (DIM / 4); i += 256) {
            int row = i / (DIM / 4), cv = i % (DIM / 4);
            float4 v = *(const float4*)(W + (size_t)(ecbase + row) * DIM + cv * 4);
            *(float4*)(&wt[row * XSTR + cv * 4]) = v;
        }
        if (c + 1 < DIM / 32) {  // hint next W tile toward cache (global_prefetch_b8)
            __builtin_prefetch(W + (size_t)(ecbase + 32 + (tid & 31)) * DIM + (tid >> 5) * 96, 0, 1);
        }
        __syncthreads();

        // 16x16 u-tile = x[rt*16.., :] @ W[ec..ec+16, :]^T via V_WMMA_F32_16X16X4_F32
        v8f acc0 = {};
        v8f acc1 = {};
        const int brow = co * 16 + lm;
#pragma unroll 4
        for (int kk = 0; kk < DIM / 4; kk += 2) {
            v2f a0 = *(const v2f*)(&xs[arow * XSTR + kk * 4 + 2 * half]);
            v2f b0 = *(const v2f*)(&wt[brow * XSTR + kk * 4 + 2 * half]);
            acc0 = __builtin_amdgcn_wmma_f32_16x16x4_f32(false, a0, false, b0,
                                                         (short)0, acc0, false, false);
            v2f a1 = *(const v2f*)(&xs[arow * XSTR + (kk + 1) * 4 + 2 * half]);
            v2f b1 = *(const v2f*)(&wt[brow * XSTR + (kk + 1) * 4 + 2 * half]);
            acc1 = __builtin_amdgcn_wmma_f32_16x16x4_f32(false, a1, false, b1,
                                                         (short)0, acc1, false, false);
        }

        // prototype values for this column (contiguous 64B per k, L0/L2 cached)
        float pv[KP];
#pragma unroll
        for (int k = 0; k < KP; ++k) pv[k] = P[k * DIM + ec + lm];

        const float bia = bl[ec + lm];
#pragma unroll
        for (int j = 0; j < 8; ++j) {
            float uv = fast_tanh(acc0[j] + acc1[j] + bia);  // u[row, ec+lm]
#pragma unroll
            for (int k = 0; k < KP; ++k) {
                float d = uv - pv[k];
                ca[j][k] = fmaf(d, d, ca[j][k]);
            }
        }
    }

    // ---- reduce squared distances across the 16 lanes of each half ----
#pragma unroll
    for (int j = 0; j < 8; ++j)
#pragma unroll
        for (int k = 0; k < KP; ++k) {
            float v = ca[j][k];
            v += __shfl_xor(v, 1, 32);
            v += __shfl_xor(v, 2, 32);
            v += __shfl_xor(v, 4, 32);
            v += __shfl_xor(v, 8, 32);
            ca[j][k] = v;
        }
    // two-phase deterministic combine of the two column-half partials
    if (co == 0 && lm == 0) {
#pragma unroll
        for (int j = 0; j < 8; ++j)
#pragma unroll
            for (int k = 0; k < KP; ++k)
                costs[(rt * 16 + j + 8 * half) * KP + k] = ca[j][k];
    }
    __syncthreads();
    if (co == 1 && lm == 0) {
#pragma unroll
        for (int j = 0; j < 8; ++j)
#pragma unroll
            for (int k = 0; k < KP; ++k)
                costs[(rt * 16 + j + 8 * half) * KP + k] += ca[j][k];
    }
    __syncthreads();

    // ---- log-domain Sinkhorn on [64 x 8] ----
    float lk[KP];
    if (tid < NWORD) {
#pragma unroll
        for (int k = 0; k < KP; ++k)
            lk[k] = -sqrtf(costs[tid * KP + k]) * (1.0f / EPSI);
    }
    for (int it = 0; it < NIT; ++it) {
        if (tid < NWORD) {  // row norm over K (registers)
            float m = lk[0];
#pragma unroll
            for (int k = 1; k < KP; ++k) m = fmaxf(m, lk[k]);
            float s = 0.0f;
#pragma unroll
            for (int k = 0; k < KP; ++k) s += __expf(lk[k] - m);
            float lse = m + __logf(s);
#pragma unroll
            for (int k = 0; k < KP; ++k) {
                lk[k] -= lse;
                costs[tid * KP + k] = lk[k];
            }
        }
        __syncthreads();
        if (tid < KP) {  // col norm over N (LDS)
            float m = -3.4e38f;
            for (int n = 0; n < NWORD; ++n) m = fmaxf(m, costs[n * KP + tid]);
            float s = 0.0f;
            for (int n = 0; n < NWORD; ++n) s += __expf(costs[n * KP + tid] - m);
            colL[tid] = m + __logf(s);
        }
        __syncthreads();
        if (tid < NWORD) {
#pragma unroll
            for (int k = 0; k < KP; ++k) lk[k] -= colL[k];
        }
    }

    if (tid < NWORD) {  // attn = mean over K of exp(logK)
        float s = 0.0f;
#pragma unroll
        for (int k = 0; k < KP; ++k) s += __expf(lk[k]);
        float a = s * (1.0f / KP);
        attnl[tid] = a;
        Out[(size_t)BATCH * DIM + (size_t)b * NWORD + tid] = a;
    }
    __syncthreads();

    // ---- weighted pooling from the resident LDS x tile ----
#pragma unroll
    for (int cc = 0; cc < DIM / 256; ++cc) {
        int d = tid + cc * 256;
        float acc = 0.0f;
        for (int n = 0; n < NWORD; ++n)
            acc = fmaf(xs[n * XSTR + d], attnl[n], acc);
        Out[(size_t)b * DIM + d] = acc;
    }
}

extern "C" void kernel_launch(void* const* d_in, const int* in_sizes, int n_in,
                              void* d_out, int out_size, void* d_ws, size_t ws_size,
                              hipStream_t stream) {
    const float* x  = (const float*)d_in[0];  // [B, N, D]
    const float* W  = (const float*)d_in[1];  // [D, D]
    const float* bv = (const float*)d_in[2];  // [D]
    const float* P  = (const float*)d_in[3];  // [K, D]
    float* out = (float*)d_out;               // s [B*D] ++ attn [B*N]

    ot_word_attn_fused<<<BATCH, 256, 0, stream>>>(x, W, bv, P, out);
}